// FactorisedRGCN_88648124990707
// MI455X (gfx1250) — compile-verified
//
#include <hip/hip_runtime.h>
#include <hip/hip_bf16.h>
#include <cstddef>

// ---------------- problem constants (match reference) ----------------
#define NN      100000
#define EE      1600000
#define NRELC   3
#define GG      512
#define DIMC    64
#define HIDC    96
#define NPOSC   50
#define EPSC    1e-5f

typedef float v2f __attribute__((ext_vector_type(2)));
typedef float v8f __attribute__((ext_vector_type(8)));

// ---------------------------------------------------------------------
// Node feature init: x[n][d] = shape_emb + col_emb + pos_emb (clamped)
// ---------------------------------------------------------------------
__global__ __launch_bounds__(256) void node_init_kernel(
    const int* __restrict__ sid, const int* __restrict__ cid,
    const int* __restrict__ pid,
    const float* __restrict__ se, const float* __restrict__ ce,
    const float* __restrict__ pe, float* __restrict__ X)
{
    int i = blockIdx.x * 256 + threadIdx.x;
    if (i >= NN * DIMC) return;
    int n = i >> 6;            // /64
    int d = i & 63;
    int p = pid[n]; p = p < 0 ? 0 : (p > NPOSC - 1 ? NPOSC - 1 : p);
    X[i] = se[sid[n] * DIMC + d] + ce[cid[n] * DIMC + d] + pe[p * DIMC + d];
}

// ---------------------------------------------------------------------
// Edge degree count per (dst, rel), then reciprocal
// ---------------------------------------------------------------------
__global__ __launch_bounds__(256) void edge_count_kernel(
    const int* __restrict__ ei, const int* __restrict__ et,
    float* __restrict__ cnt)
{
    int e = blockIdx.x * 256 + threadIdx.x;
    if (e >= EE) return;
    atomicAdd(&cnt[(size_t)ei[EE + e] * NRELC + et[e]], 1.0f);
}

__global__ __launch_bounds__(256) void inv_kernel(float* __restrict__ cnt)
{
    int i = blockIdx.x * 256 + threadIdx.x;
    if (i >= NN * NRELC) return;
    float c = cnt[i];
    cnt[i] = (c > 0.0f) ? (1.0f / c) : 0.0f;
}

// ---------------------------------------------------------------------
// Edge scatter: agg[(dst*3+t)*D + d] += x[src*D + d] * inv[dst*3+t]
// One wave32 per edge; lanes stride the feature dim.
// ---------------------------------------------------------------------
__global__ __launch_bounds__(256) void edge_scatter_kernel(
    const int* __restrict__ ei, const int* __restrict__ et,
    const float* __restrict__ inv, const float* __restrict__ X,
    float* __restrict__ Agg, int D)
{
    int wv   = (blockIdx.x * 256 + threadIdx.x) >> 5;
    int lane = threadIdx.x & 31;
    if (wv >= EE) return;
    int src = ei[wv];
    int dst = ei[EE + wv];
    int t   = et[wv];
    float w = inv[(size_t)dst * NRELC + t];
    const float* xs = X + (size_t)src * D;
    float* ag = Agg + ((size_t)dst * NRELC + t) * D;
    for (int d = lane; d < D; d += 32)
        atomicAdd(&ag[d], xs[d] * w);
}

// ---------------------------------------------------------------------
// Pack [W_root ; W_rel(0..2)] -> WMMA-B-fragment-swizzled buffer.
// Wsw[(kstep*6 + t)*32 + lane] = float2 that `lane` feeds to
// V_WMMA_F32_16X16X4_F32 for k-step `kstep`, column tile `t`:
//   rows k+khalf, k+khalf+1 of W, column t*16 + (lane&15),
//   khalf = (lane>>4)*2.  One coalesced b64 load per fragment.
// ---------------------------------------------------------------------
__global__ __launch_bounds__(256) void pack_w_sw_kernel(
    const float* __restrict__ Wroot, const float* __restrict__ Wrel,
    v2f* __restrict__ Wsw, int rootK, int K)
{
    int idx = blockIdx.x * 256 + threadIdx.x;
    int total = (K / 4) * 6 * 32;
    if (idx >= total) return;
    int kstep = idx / 192;          // 6 tiles * 32 lanes
    int rem   = idx % 192;
    int t     = rem / 32;
    int L     = rem % 32;
    int rr    = L & 15;
    int khalf = (L >> 4) << 1;
    int k     = kstep * 4 + khalf;  // rootK multiple of 4 -> k,k+1 same side
    int col   = t * 16 + rr;
    float w0, w1;
    if (k < rootK) {
        w0 = Wroot[(size_t)k * HIDC + col];
        w1 = Wroot[(size_t)(k + 1) * HIDC + col];
    } else {
        w0 = Wrel[(size_t)(k - rootK) * HIDC + col];
        w1 = Wrel[(size_t)(k + 1 - rootK) * HIDC + col];
    }
    Wsw[idx] = (v2f){w0, w1};
}

// ---------------------------------------------------------------------
// WMMA f32 GEMM:  Y[n][h] = bias[h] + [Aroot | Aagg](n,:) @ W
// One wave per 32-row slab: two 16-row C tiles x 6 column tiles
// (12 v8f accumulators). Each B fragment (one b64 load/lane) feeds
// two WMMAs. K loop split root/agg so address math is branch-free.
// ---------------------------------------------------------------------
__global__ __launch_bounds__(32) void rgcn_gemm_wmma_kernel(
    const float* __restrict__ Aroot, int rootK,
    const float* __restrict__ Aagg,  int aggK,
    const v2f* __restrict__ Wsw,          // [(K/4)*6*32] swizzled fragments
    const float* __restrict__ bias,       // [96]
    float* __restrict__ Y)                // [N][96]
{
    const int row0 = blockIdx.x * 32;
    if (row0 >= NN) return;               // uniform across the wave
    const int lane  = threadIdx.x;        // 0..31
    const int rr    = lane & 15;          // A row / B col within tile
    const int khalf = (lane >> 4) << 1;   // 0 or 2

    const float* Ar0 = Aroot + (size_t)(row0 + rr) * rootK + khalf;
    const float* Ar1 = Aroot + (size_t)(row0 + 16 + rr) * rootK + khalf;
    const float* Ag0 = Aagg  + (size_t)(row0 + rr) * aggK + khalf;
    const float* Ag1 = Aagg  + (size_t)(row0 + 16 + rr) * aggK + khalf;

    v8f acc[12];
#pragma unroll
    for (int t = 0; t < 12; ++t)
        acc[t] = (v8f){0.f,0.f,0.f,0.f,0.f,0.f,0.f,0.f};

    const v2f* wp = Wsw + lane;
    int kstep = 0;

    // ---- root-feature phase ----
    for (int k = 0; k < rootK; k += 4, ++kstep) {
        v2f a0 = *(const v2f*)(Ar0 + k);
        v2f a1 = *(const v2f*)(Ar1 + k);
        const v2f* wb = wp + (size_t)kstep * 192;
        __builtin_prefetch(wb + 2 * 192, 0, 0);   // global_prefetch_b8
#pragma unroll
        for (int t = 0; t < 6; ++t) {
            v2f b = wb[t * 32];
            acc[t]     = __builtin_amdgcn_wmma_f32_16x16x4_f32(
                false, a0, false, b, (short)0, acc[t],     false, false);
            acc[6 + t] = __builtin_amdgcn_wmma_f32_16x16x4_f32(
                false, a1, false, b, (short)0, acc[6 + t], false, false);
        }
    }

    // ---- aggregated-neighbor phase ----
    for (int k = 0; k < aggK; k += 4, ++kstep) {
        v2f a0 = *(const v2f*)(Ag0 + k);
        v2f a1 = *(const v2f*)(Ag1 + k);
        const v2f* wb = wp + (size_t)kstep * 192;
        __builtin_prefetch(wb + 2 * 192, 0, 0);
#pragma unroll
        for (int t = 0; t < 6; ++t) {
            v2f b = wb[t * 32];
            acc[t]     = __builtin_amdgcn_wmma_f32_16x16x4_f32(
                false, a0, false, b, (short)0, acc[t],     false, false);
            acc[6 + t] = __builtin_amdgcn_wmma_f32_16x16x4_f32(
                false, a1, false, b, (short)0, acc[6 + t], false, false);
        }
    }

    // C/D layout: VGPR i -> row = i + 8*(lane>=16), col = t*16 + (lane&15)
    const int rowAdd = (lane >> 4) << 3;
#pragma unroll
    for (int t = 0; t < 6; ++t) {
        const int col = t * 16 + rr;
        const float bc = bias[col];
#pragma unroll
        for (int i = 0; i < 8; ++i) {
            int r0 = row0 + i + rowAdd;
            Y[(size_t)r0 * HIDC + col]        = acc[t][i]     + bc;
            Y[(size_t)(r0 + 16) * HIDC + col] = acc[6 + t][i] + bc;
        }
    }
}

// ---------------------------------------------------------------------
// BatchNorm: partial sums, finalize, fused apply+ReLU
// ---------------------------------------------------------------------
__global__ __launch_bounds__(96) void bn_stats_kernel(
    const float* __restrict__ Y, float* __restrict__ stats)
{
    int c  = threadIdx.x;                 // 0..95
    int n0 = blockIdx.x * 64;
    int n1 = n0 + 64; if (n1 > NN) n1 = NN;
    float s = 0.f, s2 = 0.f;
    for (int n = n0; n < n1; ++n) {
        float y = Y[(size_t)n * HIDC + c];
        s += y; s2 += y * y;
    }
    atomicAdd(&stats[c], s);
    atomicAdd(&stats[HIDC + c], s2);
}

__global__ __launch_bounds__(96) void bn_finalize_kernel(
    const float* __restrict__ g, const float* __restrict__ be,
    float* __restrict__ stats)
{
    int c = threadIdx.x;
    if (c >= HIDC) return;
    const float invN = 1.0f / (float)NN;
    float m   = stats[c] * invN;
    float var = stats[HIDC + c] * invN - m * m;
    float sc  = g[c] * rsqrtf(var + EPSC);
    stats[2 * HIDC + c] = sc;
    stats[3 * HIDC + c] = be[c] - m * sc;
}

__global__ __launch_bounds__(256) void bn_apply_relu_kernel(
    float* __restrict__ Y, const float* __restrict__ stats)
{
    int i = blockIdx.x * 256 + threadIdx.x;
    if (i >= NN * HIDC) return;
    int c = i % HIDC;
    float y = Y[i] * stats[2 * HIDC + c] + stats[3 * HIDC + c];
    Y[i] = y > 0.f ? y : 0.f;
}

// ---------------------------------------------------------------------
// Global mean pool + linear head
// ---------------------------------------------------------------------
__global__ __launch_bounds__(256) void graph_cnt_kernel(
    const int* __restrict__ batch, float* __restrict__ gcnt)
{
    int n = blockIdx.x * 256 + threadIdx.x;
    if (n >= NN) return;
    atomicAdd(&gcnt[batch[n]], 1.0f);
}

__global__ __launch_bounds__(256) void pool_add_kernel(
    const float* __restrict__ H, const int* __restrict__ batch,
    float* __restrict__ pool)
{
    int i = blockIdx.x * 256 + threadIdx.x;
    if (i >= NN * HIDC) return;
    int n = i / HIDC, c = i % HIDC;
    atomicAdd(&pool[(size_t)batch[n] * HIDC + c], H[i]);
}

__global__ __launch_bounds__(256) void head_kernel(
    const float* __restrict__ pool, const float* __restrict__ gcnt,
    const float* __restrict__ linW, const float* __restrict__ linb,
    float* __restrict__ out)
{
    int i = blockIdx.x * 256 + threadIdx.x;
    if (i >= GG * 2) return;
    int g = i >> 1, c = i & 1;
    const float* p = pool + (size_t)g * HIDC;
    float s = 0.f;
    for (int h = 0; h < HIDC; ++h) s += p[h] * linW[h * 2 + c];
    float cg = gcnt[g];
    float ic = cg > 0.f ? 1.0f / cg : 0.0f;   // pooled/max(cnt,1): zero sum -> 0
    out[i] = s * ic + linb[c];
}

// ---------------------------------------------------------------------
// Host-side launcher
// ---------------------------------------------------------------------
extern "C" void kernel_launch(void* const* d_in, const int* in_sizes, int n_in,
                              void* d_out, int out_size, void* d_ws, size_t ws_size,
                              hipStream_t stream)
{
    (void)in_sizes; (void)n_in; (void)out_size; (void)ws_size;

    const int*   sid    = (const int*)  d_in[0];
    const int*   cid    = (const int*)  d_in[1];
    const int*   pid    = (const int*)  d_in[2];
    const int*   eidx   = (const int*)  d_in[3];   // [2, E]
    const int*   etype  = (const int*)  d_in[4];   // [E]
    const int*   batch  = (const int*)  d_in[5];   // [N]
    const float* se     = (const float*)d_in[6];
    const float* ce     = (const float*)d_in[7];
    const float* pe     = (const float*)d_in[8];
    const float* W1rel  = (const float*)d_in[9];
    const float* W1root = (const float*)d_in[10];
    const float* b1     = (const float*)d_in[11];
    const float* g1     = (const float*)d_in[12];
    const float* be1    = (const float*)d_in[13];
    const float* W2rel  = (const float*)d_in[14];
    const float* W2root = (const float*)d_in[15];
    const float* b2     = (const float*)d_in[16];
    const float* g2     = (const float*)d_in[17];
    const float* be2    = (const float*)d_in[18];
    const float* linW   = (const float*)d_in[19];
    const float* linb   = (const float*)d_in[20];
    float*       out    = (float*)d_out;

    // ---- workspace layout (floats) ----
    float* ws = (float*)d_ws;
    const size_t off_x0   = 0;                                  // N*64
    const size_t off_h1   = off_x0   + (size_t)NN * DIMC;       // N*96
    const size_t off_y2   = off_h1   + (size_t)NN * HIDC;       // N*96
    const size_t off_agg  = off_y2   + (size_t)NN * HIDC;       // N*3*96 (max)
    const size_t off_cnt  = off_agg  + (size_t)NN * NRELC * HIDC;
    const size_t off_w1   = off_cnt  + (size_t)NN * NRELC;      // (256/4)*192 v2f
    const size_t off_w2   = off_w1   + (size_t)(DIMC + 3*DIMC) * HIDC;   // (384/4)*192 v2f
    const size_t off_st   = off_w2   + (size_t)(HIDC + 3*HIDC) * HIDC;   // 4*96
    const size_t off_pool = off_st   + 4 * HIDC;                // G*96 + G
    float* x0   = ws + off_x0;
    float* h1   = ws + off_h1;
    float* y2   = ws + off_y2;
    float* agg  = ws + off_agg;
    float* cnt  = ws + off_cnt;
    v2f*   Wsw1 = (v2f*)(ws + off_w1);
    v2f*   Wsw2 = (v2f*)(ws + off_w2);
    float* st   = ws + off_st;
    float* pool = ws + off_pool;
    float* gcnt = pool + (size_t)GG * HIDC;

    const int K1 = DIMC + 3 * DIMC;   // 256
    const int K2 = HIDC + 3 * HIDC;   // 384

    // ---- zero scratch (capturable memset nodes) ----
    hipMemsetAsync(cnt,  0, (size_t)NN * NRELC * sizeof(float), stream);
    hipMemsetAsync(agg,  0, (size_t)NN * NRELC * HIDC * sizeof(float), stream);
    hipMemsetAsync(st,   0, 4 * HIDC * sizeof(float), stream);
    hipMemsetAsync(pool, 0, ((size_t)GG * HIDC + GG) * sizeof(float), stream);

    // ---- node features + weight packing (fragment-swizzled) ----
    node_init_kernel<<<(NN * DIMC + 255) / 256, 256, 0, stream>>>(
        sid, cid, pid, se, ce, pe, x0);
    pack_w_sw_kernel<<<((K1 / 4) * 192 + 255) / 256, 256, 0, stream>>>(
        W1root, W1rel, Wsw1, DIMC, K1);
    pack_w_sw_kernel<<<((K2 / 4) * 192 + 255) / 256, 256, 0, stream>>>(
        W2root, W2rel, Wsw2, HIDC, K2);

    // ---- per-(dst,rel) mean normalizers (shared by both layers) ----
    edge_count_kernel<<<(EE + 255) / 256, 256, 0, stream>>>(eidx, etype, cnt);
    inv_kernel<<<(NN * NRELC + 255) / 256, 256, 0, stream>>>(cnt);

    // ---- layer 1: scatter raw x, packed WMMA GEMM, BN+ReLU ----
    edge_scatter_kernel<<<(EE * 32 + 255) / 256, 256, 0, stream>>>(
        eidx, etype, cnt, x0, agg, DIMC);
    rgcn_gemm_wmma_kernel<<<(NN + 31) / 32, 32, 0, stream>>>(
        x0, DIMC, agg, 3 * DIMC, Wsw1, b1, h1);
    bn_stats_kernel<<<(NN + 63) / 64, 96, 0, stream>>>(h1, st);
    bn_finalize_kernel<<<1, 96, 0, stream>>>(g1, be1, st);
    bn_apply_relu_kernel<<<(NN * HIDC + 255) / 256, 256, 0, stream>>>(h1, st);

    // ---- layer 2 ----
    hipMemsetAsync(agg, 0, (size_t)NN * NRELC * HIDC * sizeof(float), stream);
    hipMemsetAsync(st,  0, 4 * HIDC * sizeof(float), stream);
    edge_scatter_kernel<<<(EE * 32 + 255) / 256, 256, 0, stream>>>(
        eidx, etype, cnt, h1, agg, HIDC);
    rgcn_gemm_wmma_kernel<<<(NN + 31) / 32, 32, 0, stream>>>(
        h1, HIDC, agg, 3 * HIDC, Wsw2, b2, y2);
    bn_stats_kernel<<<(NN + 63) / 64, 96, 0, stream>>>(y2, st);
    bn_finalize_kernel<<<1, 96, 0, stream>>>(g2, be2, st);
    bn_apply_relu_kernel<<<(NN * HIDC + 255) / 256, 256, 0, stream>>>(y2, st);

    // ---- mean pool + head ----
    graph_cnt_kernel<<<(NN + 255) / 256, 256, 0, stream>>>(batch, gcnt);
    pool_add_kernel<<<(NN * HIDC + 255) / 256, 256, 0, stream>>>(y2, batch, pool);
    head_kernel<<<(GG * 2 + 255) / 256, 256, 0, stream>>>(pool, gcnt, linW, linb, out);
}